// BipartiteHeteroGNN_3590592660124
// MI455X (gfx1250) — compile-verified
//
#include <hip/hip_runtime.h>
#include <hip/hip_bf16.h>

#define NV 20000
#define NC 20000
#define NE 500000
#define HID 64

typedef __attribute__((ext_vector_type(2))) float v2f;
typedef __attribute__((ext_vector_type(8))) float v8f;

#if defined(__has_builtin)
#if __has_builtin(__builtin_amdgcn_global_load_async_to_lds_b128)
#define USE_ASYNC_LDS 1
#endif
#endif
#ifndef USE_ASYNC_LDS
#define USE_ASYNC_LDS 0
#endif

// 16-byte int vector type matching the async-to-LDS builtin's parameter type
typedef int vi4 __attribute__((vector_size(16)));
typedef __attribute__((address_space(1))) vi4* gvec_t;
typedef __attribute__((address_space(3))) vi4* lvec_t;

// Stage 8192 floats (32KB) of weights into LDS, cooperatively by one wave.
// Uses gfx1250 async global->LDS (ASYNCcnt) when available.
__device__ __forceinline__ void stage_weights(const float* __restrict__ W,
                                              float* wl, int lane)
{
#if USE_ASYNC_LDS
    gvec_t g = (gvec_t)W;
    lvec_t l = (lvec_t)wl;
    for (int i = 0; i < 64; ++i) {
        int idx = i * 32 + lane;  // 16B-vector index; 512B per instruction/wave
        __builtin_amdgcn_global_load_async_to_lds_b128(g + idx, l + idx, 0, 0);
    }
#else
    for (int i = lane; i < 2048; i += 32)
        ((float4*)wl)[i] = ((const float4*)W)[i];
#endif
}

__device__ __forceinline__ void wait_async_lds()
{
#if USE_ASYNC_LDS
#if __has_builtin(__builtin_amdgcn_s_wait_asynccnt)
    __builtin_amdgcn_s_wait_asynccnt(0);
#else
    asm volatile("s_wait_asynccnt 0x0" ::: "memory");
#endif
#endif
}

// ---------------------------------------------------------------------------
// Encoder: out[:, :32] = relu(x @ ew + eb)
//          out[:, 32:] = relu(0.5*(pe_mlp(pe) + pe_mlp(-pe)))
// One 64-thread block per node.
// ---------------------------------------------------------------------------
__global__ __launch_bounds__(64) void encode_kernel(
    const float* __restrict__ x, const float* __restrict__ pe,
    const float* __restrict__ ew, const float* __restrict__ eb,
    const float* __restrict__ pw1, const float* __restrict__ pb1,
    const float* __restrict__ pw2, const float* __restrict__ pb2,
    float* __restrict__ out)
{
    __shared__ float h1p[64];
    __shared__ float h1m[64];
    const int node = blockIdx.x;
    const int t = threadIdx.x;

    float accp = pb1[t];
    float accm = pb1[t];
    for (int k = 0; k < 8; ++k) {
        float p = pe[node * 8 + k];
        float w = pw1[k * 64 + t];
        accp += p * w;
        accm -= p * w;
    }
    h1p[t] = fmaxf(accp, 0.f);
    h1m[t] = fmaxf(accm, 0.f);
    __syncthreads();

    if (t < 32) {
        float a = pb2[t], b = pb2[t];
        for (int j = 0; j < 64; ++j) {
            float w = pw2[j * 32 + t];
            a += h1p[j] * w;
            b += h1m[j] * w;
        }
        float pef = 0.5f * (a + b);
        float lin = eb[t] + x[node * 2 + 0] * ew[0 * 32 + t]
                          + x[node * 2 + 1] * ew[1 * 32 + t];
        out[node * HID + t]      = fmaxf(lin, 0.f);
        out[node * HID + 32 + t] = fmaxf(pef, 0.f);
    }
}

// ---------------------------------------------------------------------------
// Zero scratch accumulators (m, s, t)
// ---------------------------------------------------------------------------
__global__ void zero_kernel(float* __restrict__ p, int n)
{
    int i = blockIdx.x * blockDim.x + threadIdx.x;
    if (i < n) p[i] = 0.f;
}

// ---------------------------------------------------------------------------
// Pass 1: msg = relu(x_src[src] + ewt*lew + leb) + 1e-7 ; segment-max into m.
// msg > 0 so positive-float atomic max == signed-int atomic max on the bits.
// One thread per (edge, channel-quad): b128 gathers, 4 atomics each.
// ---------------------------------------------------------------------------
__global__ __launch_bounds__(256) void msg_max_kernel(
    const float* __restrict__ xsrc,
    const int* __restrict__ src, const int* __restrict__ dst,
    const float* __restrict__ ewt,
    const float* __restrict__ lew, const float* __restrict__ leb,
    float* __restrict__ mbuf)
{
    int idx = blockIdx.x * blockDim.x + threadIdx.x;
    if (idx >= NE * 16) return;
    int q = (idx & 15) << 2;
    int e = idx >> 4;
    if ((idx & 15) == 0 && e + 16384 < NE) {
        __builtin_prefetch(src + e + 16384, 0, 1);   // global_prefetch_b8
        __builtin_prefetch(dst + e + 16384, 0, 1);
    }
    int s = src[e];
    int d = dst[e];
    float w = ewt[e];
    const float4 xs = *(const float4*)&xsrc[s * HID + q];
    const float4 lw = *(const float4*)&lew[q];
    const float4 lb = *(const float4*)&leb[q];
    float4 m;
    m.x = fmaxf(xs.x + w * lw.x + lb.x, 0.f) + 1e-7f;
    m.y = fmaxf(xs.y + w * lw.y + lb.y, 0.f) + 1e-7f;
    m.z = fmaxf(xs.z + w * lw.z + lb.z, 0.f) + 1e-7f;
    m.w = fmaxf(xs.w + w * lw.w + lb.w, 0.f) + 1e-7f;
    int* mb = (int*)&mbuf[d * HID + q];
    atomicMax(mb + 0, __float_as_int(m.x));
    atomicMax(mb + 1, __float_as_int(m.y));
    atomicMax(mb + 2, __float_as_int(m.z));
    atomicMax(mb + 3, __float_as_int(m.w));
}

// ---------------------------------------------------------------------------
// Pass 2: recompute msg, ex = exp(msg - m[dst]); accumulate s += ex,
// t += ex*msg. (agg = t / (s + 1e-16) afterwards — exact fusion since the
// denominator is per-(dst,channel).)
// ---------------------------------------------------------------------------
__global__ __launch_bounds__(256) void exp_sum_kernel(
    const float* __restrict__ xsrc,
    const int* __restrict__ src, const int* __restrict__ dst,
    const float* __restrict__ ewt,
    const float* __restrict__ lew, const float* __restrict__ leb,
    const float* __restrict__ mbuf,
    float* __restrict__ sbuf, float* __restrict__ tbuf)
{
    int idx = blockIdx.x * blockDim.x + threadIdx.x;
    if (idx >= NE * 16) return;
    int q = (idx & 15) << 2;
    int e = idx >> 4;
    int s = src[e];
    int d = dst[e];
    float w = ewt[e];
    const float4 xs = *(const float4*)&xsrc[s * HID + q];
    const float4 lw = *(const float4*)&lew[q];
    const float4 lb = *(const float4*)&leb[q];
    const float4 mv = *(const float4*)&mbuf[d * HID + q];
    float msg0 = fmaxf(xs.x + w * lw.x + lb.x, 0.f) + 1e-7f;
    float msg1 = fmaxf(xs.y + w * lw.y + lb.y, 0.f) + 1e-7f;
    float msg2 = fmaxf(xs.z + w * lw.z + lb.z, 0.f) + 1e-7f;
    float msg3 = fmaxf(xs.w + w * lw.w + lb.w, 0.f) + 1e-7f;
    float ex0 = expf(msg0 - mv.x);
    float ex1 = expf(msg1 - mv.y);
    float ex2 = expf(msg2 - mv.z);
    float ex3 = expf(msg3 - mv.w);
    float* sb = &sbuf[d * HID + q];
    float* tb = &tbuf[d * HID + q];
    atomicAdd(sb + 0, ex0);
    atomicAdd(sb + 1, ex1);
    atomicAdd(sb + 2, ex2);
    atomicAdd(sb + 3, ex3);
    atomicAdd(tb + 0, ex0 * msg0);
    atomicAdd(tb + 1, ex1 * msg1);
    atomicAdd(tb + 2, ex2 * msg2);
    atomicAdd(tb + 3, ex3 * msg3);
}

// ---------------------------------------------------------------------------
// GENConv node MLP via fp32 WMMA (exact): h = t/(s+1e-16) + x_dst,
// out = relu(h @ W1 + b1) @ W2 + b2 with W1:[64,128], W2:[128,64].
// One wave per 16-node tile; 16x16x4 f32 WMMA tiles.
// Weights are staged through one 32KB LDS region (W1 then W2) using the
// async global->LDS path (ASYNCcnt); B fragments are ds_loads from LDS.
// A layout (ISA 7.12.2, 32-bit 16x4): lane L holds row M=L&15,
//   K = {0,1} for lanes 0-15, {2,3} for lanes 16-31 within each K-quad.
// B layout mirrors; C/D: VGPR v -> row v / v+8, col = lane&15.
// ---------------------------------------------------------------------------
__global__ __launch_bounds__(32) void genconv_mlp_kernel(
    const float* __restrict__ xdst,
    const float* __restrict__ sbuf, const float* __restrict__ tbuf,
    const float* __restrict__ W1, const float* __restrict__ B1,
    const float* __restrict__ W2, const float* __restrict__ B2,
    float* __restrict__ out)
{
    __shared__ float wl[2 * HID * HID];   // 8192 floats = 32KB (W1, then W2)
    __shared__ float h[16 * HID];         // input tile  (16 x 64)
    __shared__ float hid[16 * 2 * HID];   // hidden tile (16 x 128)

    const int lane = threadIdx.x;
    const int base = blockIdx.x * 16;
    const int ln = lane & 15;
    const int hi2 = (lane >> 4) << 1;      // 0 or 2: K sub-offset
    const int rowoff = (lane >> 4) << 3;   // 0 or 8: C/D row offset

    // Kick off async staging of W1, overlap with computing the input tile.
    stage_weights(W1, wl, lane);
    for (int i = lane; i < 16 * HID; i += 32) {
        int r = i >> 6, c = i & (HID - 1);
        int node = base + r;
        float sv = sbuf[node * HID + c];
        float tv = tbuf[node * HID + c];
        h[i] = tv / (sv + 1e-16f) + xdst[node * HID + c];
    }
    wait_async_lds();
    __syncthreads();

    // hidden = relu(h @ W1 + b1): N = 128 in 8 tiles, K = 64 in 16 steps
    for (int nt = 0; nt < 8; ++nt) {
        int ncol = nt * 16 + ln;
        float bias = B1[ncol];
        v8f acc = {bias, bias, bias, bias, bias, bias, bias, bias};
        for (int k = 0; k < 16; ++k) {
            int kk = k * 4 + hi2;
            v2f a = {h[ln * HID + kk], h[ln * HID + kk + 1]};
            v2f b = {wl[kk * 128 + ncol], wl[(kk + 1) * 128 + ncol]};
            acc = __builtin_amdgcn_wmma_f32_16x16x4_f32(
                false, a, false, b, (short)0, acc, false, false);
        }
        for (int v = 0; v < 8; ++v)
            hid[(v + rowoff) * 128 + ncol] = fmaxf(acc[v], 0.f);
    }

    // All DS reads of wl must retire before the async engine overwrites it.
    asm volatile("s_wait_dscnt 0x0" ::: "memory");
    stage_weights(W2, wl, lane);
    wait_async_lds();
    __syncthreads();

    // out = hidden @ W2 + b2: N = 64 in 4 tiles, K = 128 in 32 steps
    for (int nt = 0; nt < 4; ++nt) {
        int ncol = nt * 16 + ln;
        float bias = B2[ncol];
        v8f acc = {bias, bias, bias, bias, bias, bias, bias, bias};
        for (int k = 0; k < 32; ++k) {
            int kk = k * 4 + hi2;
            v2f a = {hid[ln * 128 + kk], hid[ln * 128 + kk + 1]};
            v2f b = {wl[kk * HID + ncol], wl[(kk + 1) * HID + ncol]};
            acc = __builtin_amdgcn_wmma_f32_16x16x4_f32(
                false, a, false, b, (short)0, acc, false, false);
        }
        for (int v = 0; v < 8; ++v)
            out[(base + v + rowoff) * HID + ncol] = acc[v];
    }
}

// ---------------------------------------------------------------------------
// Prediction head: out[node*3 + layer] = relu(f @ W1 + b1) @ W2 + b2
// One 64-thread block per node.
// ---------------------------------------------------------------------------
__global__ __launch_bounds__(64) void pred_kernel(
    const float* __restrict__ feat,
    const float* __restrict__ W1, const float* __restrict__ B1,
    const float* __restrict__ W2, const float* __restrict__ B2,
    float* __restrict__ out, int layer)
{
    __shared__ float red[64];
    const int node = blockIdx.x;
    const int t = threadIdx.x;
    const float* f = feat + node * HID;
    float acc = B1[t];
    for (int j = 0; j < HID; ++j)
        acc += f[j] * W1[j * HID + t];
    red[t] = fmaxf(acc, 0.f) * W2[t];
    __syncthreads();
    for (int s = 32; s > 0; s >>= 1) {
        if (t < s) red[t] += red[t + s];
        __syncthreads();
    }
    if (t == 0) out[node * 3 + layer] = red[0] + B2[0];
}

// ---------------------------------------------------------------------------
// Host-side conv driver
// ---------------------------------------------------------------------------
static void run_conv(const float* x_src, const float* x_dst,
                     const int* src, const int* dst, const float* ewt,
                     const float* lew, const float* leb,
                     const float* W1, const float* B1,
                     const float* W2, const float* B2,
                     float* mbuf, float* sbuf, float* tbuf,
                     float* out, int n_dst, hipStream_t stream)
{
    int nz = 3 * 20000 * HID;  // m, s, t are contiguous
    zero_kernel<<<(nz + 255) / 256, 256, 0, stream>>>(mbuf, nz);
    int nthr = NE * 16;
    msg_max_kernel<<<(nthr + 255) / 256, 256, 0, stream>>>(
        x_src, src, dst, ewt, lew, leb, mbuf);
    exp_sum_kernel<<<(nthr + 255) / 256, 256, 0, stream>>>(
        x_src, src, dst, ewt, lew, leb, mbuf, sbuf, tbuf);
    genconv_mlp_kernel<<<n_dst / 16, 32, 0, stream>>>(
        x_dst, sbuf, tbuf, W1, B1, W2, B2, out);
}

extern "C" void kernel_launch(void* const* d_in, const int* in_sizes, int n_in,
                              void* d_out, int out_size, void* d_ws, size_t ws_size,
                              hipStream_t stream)
{
    const float* x_vals  = (const float*)d_in[0];
    const float* x_cons  = (const float*)d_in[1];
    const float* pe_vals = (const float*)d_in[2];
    const float* pe_cons = (const float*)d_in[3];
    const int*   ei_v2c  = (const int*)d_in[4];   // [2, E]: row0 src(vals), row1 dst(cons)
    const int*   ei_c2v  = (const int*)d_in[5];
    const float* ew_v2c  = (const float*)d_in[6];
    const float* ew_c2v  = (const float*)d_in[7];

    const float* enc_vals_w = (const float*)d_in[8];
    const float* enc_vals_b = (const float*)d_in[9];
    const float* pe_vals_w1 = (const float*)d_in[10];
    const float* pe_vals_b1 = (const float*)d_in[11];
    const float* pe_vals_w2 = (const float*)d_in[12];
    const float* pe_vals_b2 = (const float*)d_in[13];
    const float* enc_cons_w = (const float*)d_in[14];
    const float* enc_cons_b = (const float*)d_in[15];
    const float* pe_cons_w1 = (const float*)d_in[16];
    const float* pe_cons_b1 = (const float*)d_in[17];
    const float* pe_cons_w2 = (const float*)d_in[18];
    const float* pe_cons_b2 = (const float*)d_in[19];

    const float* v2c_edge_w = (const float*)d_in[20];  // [3,1,64]
    const float* v2c_edge_b = (const float*)d_in[21];  // [3,64]
    const float* v2c_w1     = (const float*)d_in[22];  // [3,64,128]
    const float* v2c_b1     = (const float*)d_in[23];  // [3,128]
    const float* v2c_w2     = (const float*)d_in[24];  // [3,128,64]
    const float* v2c_b2     = (const float*)d_in[25];  // [3,64]
    const float* c2v_edge_w = (const float*)d_in[26];
    const float* c2v_edge_b = (const float*)d_in[27];
    const float* c2v_w1     = (const float*)d_in[28];
    const float* c2v_b1     = (const float*)d_in[29];
    const float* c2v_w2     = (const float*)d_in[30];
    const float* c2v_b2     = (const float*)d_in[31];

    const float* pred_vals_w1 = (const float*)d_in[32];
    const float* pred_vals_b1 = (const float*)d_in[33];
    const float* pred_vals_w2 = (const float*)d_in[34];
    const float* pred_vals_b2 = (const float*)d_in[35];
    const float* pred_cons_w1 = (const float*)d_in[36];
    const float* pred_cons_b1 = (const float*)d_in[37];
    const float* pred_cons_w2 = (const float*)d_in[38];
    const float* pred_cons_b2 = (const float*)d_in[39];

    // Workspace layout (floats)
    float* ws    = (float*)d_ws;
    float* enc_v = ws;                       // NV*64
    float* enc_c = enc_v + (size_t)NV * HID; // NC*64
    float* hv    = enc_c + (size_t)NC * HID; // 3*NV*64
    float* hc    = hv + (size_t)3 * NV * HID;
    float* mbuf  = hc + (size_t)3 * NC * HID; // 20000*64 (m,s,t contiguous)
    float* sbuf  = mbuf + (size_t)20000 * HID;
    float* tbuf  = sbuf + (size_t)20000 * HID;

    // Encoders
    encode_kernel<<<NV, 64, 0, stream>>>(x_vals, pe_vals, enc_vals_w, enc_vals_b,
                                         pe_vals_w1, pe_vals_b1, pe_vals_w2,
                                         pe_vals_b2, enc_v);
    encode_kernel<<<NC, 64, 0, stream>>>(x_cons, pe_cons, enc_cons_w, enc_cons_b,
                                         pe_cons_w1, pe_cons_b1, pe_cons_w2,
                                         pe_cons_b2, enc_c);

    const float* cur_v = enc_v;
    const float* cur_c = enc_c;
    for (int i = 0; i < 3; ++i) {
        // v2c: src = vals, dst = cons
        float* new_c = hc + (size_t)i * NC * HID;
        run_conv(cur_v, cur_c, ei_v2c, ei_v2c + NE, ew_v2c,
                 v2c_edge_w + i * 64, v2c_edge_b + i * 64,
                 v2c_w1 + i * 64 * 128, v2c_b1 + i * 128,
                 v2c_w2 + i * 128 * 64, v2c_b2 + i * 64,
                 mbuf, sbuf, tbuf, new_c, NC, stream);
        cur_c = new_c;
        // c2v: src = cons, dst = vals
        float* new_v = hv + (size_t)i * NV * HID;
        run_conv(cur_c, cur_v, ei_c2v, ei_c2v + NE, ew_c2v,
                 c2v_edge_w + i * 64, c2v_edge_b + i * 64,
                 c2v_w1 + i * 64 * 128, c2v_b1 + i * 128,
                 c2v_w2 + i * 128 * 64, c2v_b2 + i * 64,
                 mbuf, sbuf, tbuf, new_v, NV, stream);
        cur_v = new_v;
    }

    // Prediction heads: d_out = pv [NV,3] then pc [NC,3]
    float* out_v = (float*)d_out;
    float* out_c = out_v + (size_t)NV * 3;
    for (int i = 0; i < 3; ++i) {
        pred_kernel<<<NV, 64, 0, stream>>>(hv + (size_t)i * NV * HID,
                                           pred_vals_w1, pred_vals_b1,
                                           pred_vals_w2, pred_vals_b2, out_v, i);
        pred_kernel<<<NC, 64, 0, stream>>>(hc + (size_t)i * NC * HID,
                                           pred_cons_w1, pred_cons_b1,
                                           pred_cons_w2, pred_cons_b2, out_c, i);
    }
}